// TGCNCell_27668179321238
// MI455X (gfx1250) — compile-verified
//
#include <hip/hip_runtime.h>
#include <math.h>

// ---------------------------------------------------------------------------
// T-GCN cell for MI455X (gfx1250).
// GEMMs use V_WMMA_F32_16X16X4_F32 (fp32-exact matrix path, wave32).
// Edge gather/scatter uses global_atomic_add_f32 (memory-bound phase).
// N = 100000 is a multiple of 16 -> every wave owns a full 16x16 tile and
// EXEC stays all-ones across the WMMA (ISA 7.12 requirement).
// ---------------------------------------------------------------------------

typedef __attribute__((ext_vector_type(2))) float v2f;
typedef __attribute__((ext_vector_type(8))) float v8f;

#define DH 64   // hidden/feature dim (reference: D_IN = D_H = 64)

// ---- degrees -------------------------------------------------------------

__global__ void degree_count_kernel(const int* __restrict__ src,
                                    const int* __restrict__ dst,
                                    float* __restrict__ cnt_out,
                                    float* __restrict__ cnt_in, int E) {
    int e = blockIdx.x * blockDim.x + threadIdx.x;
    if (e < E) {
        atomicAdd(&cnt_out[src[e]], 1.0f);
        atomicAdd(&cnt_in[dst[e]], 1.0f);
    }
}

__global__ void degree_finalize_kernel(float* __restrict__ a,
                                       float* __restrict__ b, int N) {
    int i = blockIdx.x * blockDim.x + threadIdx.x;
    if (i < N) {
        a[i] = rsqrtf(fmaxf(a[i], 1.0f));
        b[i] = rsqrtf(fmaxf(b[i], 1.0f));
    }
}

// ---- GEMM: out[N x FOUT] = (inv_deg[n] * [A0 | A1])[N x 128] @ W[128 x FOUT]
// One wave per 16x16 output tile; K looped in steps of 4 via f32 WMMA.
// A-fragment (16x4 f32, 2 VGPR): lane L<16 holds row L, K = k0+{0,1};
// lanes 16..31 hold row L-16, K = k0+{2,3}. B mirrors with N across lanes.
// C/D (8 VGPR): M = j + 8*(lane>=16), N = lane&15.

template <int FOUT>
__global__ __launch_bounds__(256) void gemm_norm_wmma(
        const float* __restrict__ A0,      // [N x 64] first half of concat
        const float* __restrict__ A1,      // [N x 64] second half of concat
        const float* __restrict__ W,       // [128 x FOUT] row-major
        const float* __restrict__ inv_deg, // [N]
        float* __restrict__ out,           // [N x FOUT]
        int mtiles) {
    constexpr int NT = FOUT / 16;
    int tile = blockIdx.x * 8 + threadIdx.y;   // 8 waves / block
    int mt = tile / NT;
    int nt = tile % NT;
    if (mt >= mtiles) return;                  // uniform per wave

    int lane = threadIdx.x;                    // 0..31
    int half = lane >> 4;                      // 0 or 1
    int l15  = lane & 15;

    int row = mt * 16 + l15;                   // A row owned by this lane
    float s = inv_deg[row];
    const float* xr = A0 + (size_t)row * DH;
    const float* hr = A1 + (size_t)row * DH;
    int col = nt * 16 + l15;                   // B/C column owned by this lane

    v8f acc = {};
#pragma unroll
    for (int k0 = 0; k0 < 2 * DH; k0 += 4) {
        int kb = k0 + half * 2;                // kb even; 64-boundary safe
        const float* ap = (k0 < DH) ? (xr + kb) : (hr + (kb - DH));
        v2f a, b;
        a.x = ap[0] * s;
        a.y = ap[1] * s;
        b.x = W[(size_t)kb * FOUT + col];
        b.y = W[(size_t)(kb + 1) * FOUT + col];
        acc = __builtin_amdgcn_wmma_f32_16x16x4_f32(
            /*neg_a=*/false, a, /*neg_b=*/false, b,
            /*c_mod=*/(short)0, acc, /*reuse_a=*/false, /*reuse_b=*/false);
    }

#pragma unroll
    for (int j = 0; j < 8; ++j) {
        int r = mt * 16 + half * 8 + j;
        out[(size_t)r * FOUT + col] = acc[j];
    }
}

// ---- edge gather + scatter-add: agg[dst[e]][f] += xw[src[e]][f] ----------

template <int F>
__global__ void scatter_add_kernel(const float* __restrict__ xw,
                                   const int* __restrict__ src,
                                   const int* __restrict__ dst,
                                   float* __restrict__ agg, int E) {
    unsigned idx = blockIdx.x * blockDim.x + threadIdx.x;  // < E*F <= 204.8M
    unsigned total = (unsigned)E * (unsigned)F;
    if (idx >= total) return;
    unsigned e = idx / F;
    unsigned f = idx % F;
    float v = xw[(size_t)src[e] * F + f];
    atomicAdd(&agg[(size_t)dst[e] * F + f], v);
}

// ---- gates: r,u = sigmoid(agg*inv_di + b); emit u and r*h ----------------

__global__ void gates_kernel(const float* __restrict__ agg,     // [N x 128]
                             const float* __restrict__ inv_di,  // [N]
                             const float* __restrict__ b_gate,  // [128]
                             const float* __restrict__ h,       // [N x 64]
                             float* __restrict__ u,             // [N x 64]
                             float* __restrict__ rh,            // [N x 64]
                             int N) {
    int idx = blockIdx.x * blockDim.x + threadIdx.x;
    if (idx >= N * DH) return;
    int n = idx >> 6;
    int j = idx & (DH - 1);
    float inv = inv_di[n];
    float gr = agg[(size_t)n * 2 * DH + j] * inv + b_gate[j];
    float gu = agg[(size_t)n * 2 * DH + DH + j] * inv + b_gate[DH + j];
    float r  = 1.0f / (1.0f + __expf(-gr));
    float uu = 1.0f / (1.0f + __expf(-gu));
    u[idx]  = uu;
    rh[idx] = r * h[idx];
}

// ---- final: new_h = u*h + (1-u)*tanh(agg2*inv_di + b_cand) ---------------

__global__ void final_kernel(const float* __restrict__ agg2,   // [N x 64]
                             const float* __restrict__ inv_di, // [N]
                             const float* __restrict__ b_cand, // [64]
                             const float* __restrict__ h,      // [N x 64]
                             const float* __restrict__ u,      // [N x 64]
                             float* __restrict__ out, int N) {
    int idx = blockIdx.x * blockDim.x + threadIdx.x;
    if (idx >= N * DH) return;
    int n = idx >> 6;
    int j = idx & (DH - 1);
    float cand = agg2[idx] * inv_di[n] + b_cand[j];
    float uu = u[idx];
    out[idx] = uu * h[idx] + (1.0f - uu) * tanhf(cand);
}

// ---------------------------------------------------------------------------

extern "C" void kernel_launch(void* const* d_in, const int* in_sizes, int n_in,
                              void* d_out, int out_size, void* d_ws, size_t ws_size,
                              hipStream_t stream) {
    const float* x      = (const float*)d_in[0];
    const float* h      = (const float*)d_in[1];
    const int*   src    = (const int*)d_in[2];
    const int*   dst    = (const int*)d_in[3];
    const float* W_gate = (const float*)d_in[4];  // [128 x 128]
    const float* b_gate = (const float*)d_in[5];  // [128]
    const float* W_cand = (const float*)d_in[6];  // [128 x 64]
    const float* b_cand = (const float*)d_in[7];  // [64]

    const int N = in_sizes[0] / DH;   // 100000 (multiple of 16)
    const int E = in_sizes[2];        // 1600000

    // workspace layout (floats)
    float* p      = (float*)d_ws;
    float* inv_do = p;                          // N
    float* inv_di = p + N;                      // N
    float* xw     = p + 2 * (size_t)N;          // N*128 (reused N*64 in phase 2)
    float* agg    = xw + (size_t)N * 2 * DH;    // N*128 (reused N*64 in phase 2)
    float* u      = agg + (size_t)N * 2 * DH;   // N*64
    float* rh     = u + (size_t)N * DH;         // N*64
    float* out    = (float*)d_out;              // N*64

    const int mtiles = N / 16;

    // degrees
    hipMemsetAsync(inv_do, 0, sizeof(float) * 2 * (size_t)N, stream);
    degree_count_kernel<<<(E + 255) / 256, 256, 0, stream>>>(src, dst, inv_do, inv_di, E);
    degree_finalize_kernel<<<(N + 255) / 256, 256, 0, stream>>>(inv_do, inv_di, N);

    // gate conv: xw = (inv_do * [x,h]) @ W_gate ; agg = segment_sum(xw[src], dst)
    hipMemsetAsync(agg, 0, sizeof(float) * (size_t)N * 2 * DH, stream);
    {
        int tiles = mtiles * (2 * DH / 16);
        gemm_norm_wmma<2 * DH><<<(tiles + 7) / 8, dim3(32, 8), 0, stream>>>(
            x, h, W_gate, inv_do, xw, mtiles);
    }
    {
        unsigned total = (unsigned)E * (2 * DH);
        scatter_add_kernel<2 * DH><<<(total + 255) / 256, 256, 0, stream>>>(
            xw, src, dst, agg, E);
    }
    gates_kernel<<<(N * DH + 255) / 256, 256, 0, stream>>>(agg, inv_di, b_gate, h, u, rh, N);

    // candidate conv: reuse xw/agg buffers (stream order guarantees safety)
    hipMemsetAsync(agg, 0, sizeof(float) * (size_t)N * DH, stream);
    {
        int tiles = mtiles * (DH / 16);
        gemm_norm_wmma<DH><<<(tiles + 7) / 8, dim3(32, 8), 0, stream>>>(
            x, rh, W_cand, inv_do, xw, mtiles);
    }
    {
        unsigned total = (unsigned)E * DH;
        scatter_add_kernel<DH><<<(total + 255) / 256, 256, 0, stream>>>(
            xw, src, dst, agg, E);
    }
    final_kernel<<<(N * DH + 255) / 256, 256, 0, stream>>>(agg, inv_di, b_cand, h, u, out, N);
}